// CNOT_56573309222982
// MI455X (gfx1250) — compile-verified
//
#include <hip/hip_runtime.h>
#include <hip/hip_bf16.h>

// ---------------------------------------------------------------------------
// CNOT permutation: out[lin(i), :] = x[i, :], d=2, n=24, batch=4 (fp32).
// Pure HBM-streaming problem: 512 MB moved -> ~22 us floor at 23.3 TB/s.
// Strategy: CDNA5 async global->LDS->global b128 copies (ASYNCcnt-tracked),
// double-buffered 8-deep per wave so each wave keeps ~8 KB in flight and the
// store of batch k overlaps the load of batch k+1.
// ---------------------------------------------------------------------------

#define AS1 __attribute__((address_space(1)))
#define AS3 __attribute__((address_space(3)))

typedef int v4i __attribute__((ext_vector_type(4)));

__device__ __forceinline__ void async_load_b128(const void* gsrc, void* ldst) {
#if __has_builtin(__builtin_amdgcn_global_load_async_to_lds_b128)
    __builtin_amdgcn_global_load_async_to_lds_b128(
        (AS1 v4i*)(gsrc), (AS3 v4i*)(ldst), 0, 0);
#else
    AS3 char*       lp = (AS3 char*)(ldst);
    AS1 const char* gp = (AS1 const char*)(gsrc);
    asm volatile("global_load_async_to_lds_b128 %0, %1, off"
                 :: "v"(lp), "v"(gp) : "memory");
#endif
}

__device__ __forceinline__ void async_store_b128(void* gdst, const void* lsrc) {
#if __has_builtin(__builtin_amdgcn_global_store_async_from_lds_b128)
    __builtin_amdgcn_global_store_async_from_lds_b128(
        (AS1 v4i*)(gdst), (AS3 v4i*)(lsrc), 0, 0);
#else
    AS3 const char* lp = (AS3 const char*)(lsrc);
    AS1 char*       gp = (AS1 char*)(gdst);
    asm volatile("global_store_async_from_lds_b128 %0, %1, off"
                 :: "v"(gp), "v"(lp) : "memory");
#endif
}

__device__ __forceinline__ void wait_async0() {
#if __has_builtin(__builtin_amdgcn_s_wait_asynccnt)
    __builtin_amdgcn_s_wait_asynccnt(0);
#else
    asm volatile("s_wait_asynccnt 0x0" ::: "memory");
#endif
}

// lin(idx) per the reference: pt = d^(n-1-target), pc = d^(n-1-control),
// dt = (idx/pt)%d, dc = (idx/pc)%d, lin = idx + (((dt+dc)%d) - dt)*pt.
// d==2 specializes to a conditional bit flip: lin = idx ^ (((idx>>sc)&1)<<st).
__device__ __forceinline__ unsigned lin_map(unsigned idx, int d, int st, int sc,
                                            unsigned pt, unsigned pc) {
    if (d == 2) {
        unsigned dc = (idx >> sc) & 1u;
        return idx ^ (dc << st);
    }
    unsigned dt = (idx / pt) % (unsigned)d;
    unsigned dc = (idx / pc) % (unsigned)d;
    int diff = (int)((dt + dc) % (unsigned)d) - (int)dt;
    return (unsigned)((int)idx + diff * (int)pt);
}

#define STAGES 8  // async b128 ops in flight per thread per phase

__global__ __launch_bounds__(256)
void cnot_perm_async_kernel(const float* __restrict__ x,
                            const int* __restrict__ ctrlp,
                            const int* __restrict__ tgtp,
                            const int* __restrict__ dp,
                            const int* __restrict__ np,
                            float* __restrict__ out,
                            unsigned int total_elems) {
    const int d = dp[0], n = np[0];
    const int control = ctrlp[0], target = tgtp[0];
    const int st = n - 1 - target;
    const int sc = n - 1 - control;

    unsigned Dn = 1;
    for (int i = 0; i < n; ++i) Dn *= (unsigned)d;
    unsigned pt = 1;
    for (int i = 0; i < st; ++i) pt *= (unsigned)d;
    unsigned pc = 1;
    for (int i = 0; i < sc; ++i) pc *= (unsigned)d;
    const unsigned batch = (Dn > 0) ? (total_elems / Dn) : 0;

    const unsigned tid   = threadIdx.x;
    const unsigned gsize = blockDim.x * gridDim.x;
    const unsigned g0    = blockIdx.x * blockDim.x + tid;

    if (batch == 4 && ((total_elems & 3u) == 0u)) {
        // ---- fast path: one b128 per row, async-pipelined through LDS ----
        __shared__ float4 buf[2][STAGES][256];   // 64 KB / block
        const float4* x4 = (const float4*)x;
        float4*       o4 = (float4*)out;
        const unsigned nrows = Dn;

        // Uniform super-iteration count (same for all waves; per-op guards
        // handle the ragged tail).
        const unsigned per_thread = (nrows + gsize - 1) / gsize;
        const unsigned niter      = (per_thread + STAGES - 1) / STAGES;

        // Prologue: issue loads for batch 0 into buf[0].
        #pragma unroll
        for (int k = 0; k < STAGES; ++k) {
            unsigned long long row = (unsigned long long)g0 +
                                     (unsigned long long)k * gsize;
            if (row < nrows)
                async_load_b128(&x4[row], &buf[0][k][tid]);
        }

        int b = 0;
        for (unsigned it = 0; it < niter; ++it) {
            // Waits for: loads(it) done (buf[b] valid) AND stores(it-1) done
            // (buf[b^1] drained) -> safe to store buf[b] and refill buf[b^1].
            wait_async0();

            #pragma unroll
            for (int k = 0; k < STAGES; ++k) {
                unsigned long long row = (unsigned long long)g0 +
                    ((unsigned long long)it * STAGES + k) * gsize;
                if (row < nrows) {
                    unsigned l = lin_map((unsigned)row, d, st, sc, pt, pc);
                    async_store_b128(&o4[l], &buf[b][k][tid]);
                }
            }
            if (it + 1 < niter) {
                #pragma unroll
                for (int k = 0; k < STAGES; ++k) {
                    unsigned long long row = (unsigned long long)g0 +
                        ((unsigned long long)(it + 1) * STAGES + k) * gsize;
                    if (row < nrows)
                        async_load_b128(&x4[row], &buf[b ^ 1][k][tid]);
                }
            }
            b ^= 1;
        }
        wait_async0();   // drain final stores before wave exit
    } else {
        // ---- generic fallback: scalar permuted copy (any d/batch) ----
        if (batch == 0) return;
        for (unsigned long long e = g0; e < (unsigned long long)total_elems;
             e += gsize) {
            unsigned row = (unsigned)(e / batch);
            unsigned col = (unsigned)(e - (unsigned long long)row * batch);
            unsigned l   = lin_map(row, d, st, sc, pt, pc);
            out[(unsigned long long)l * batch + col] = x[e];
        }
    }
}

extern "C" void kernel_launch(void* const* d_in, const int* in_sizes, int n_in,
                              void* d_out, int out_size, void* d_ws, size_t ws_size,
                              hipStream_t stream) {
    const float* x    = (const float*)d_in[0];
    const int*   ctrl = (const int*)d_in[1];
    const int*   tgt  = (const int*)d_in[2];
    const int*   dpar = (const int*)d_in[3];
    const int*   npar = (const int*)d_in[4];
    float*       out  = (float*)d_out;

    const unsigned total = (unsigned)out_size;          // 2^26 for this instance
    unsigned rows = total / 4u;                         // 2^24 b128 tiles
    if (rows == 0) rows = 1;

    // 256 thr * 8 stages * 2 super-iterations = 4096 rows per block
    unsigned blocks = (rows + 4095u) / 4096u;           // 4096 blocks here
    if (blocks < 1u) blocks = 1u;

    cnot_perm_async_kernel<<<dim3(blocks), dim3(256), 0, stream>>>(
        x, ctrl, tgt, dpar, npar, out, total);
}